// MatrixSequenceMerger_2508260901807
// MI455X (gfx1250) — compile-verified
//
#include <hip/hip_runtime.h>
#include <hip/hip_bf16.h>

typedef __attribute__((ext_vector_type(2))) float v2f;
typedef __attribute__((ext_vector_type(4))) float v4f;
typedef __attribute__((ext_vector_type(8))) float v8f;

static constexpr int kB = 4, kL = 1024, kC = 64, kD = 64, kOUT = 256, kK = 2 * kD + kC; // 192

// ---------------------------------------------------------------------------
// Kernel 1: fused score (matrix . Wp + bp), row softmax, and mean over j.
// One block per (b,i). Streams 256 KB of matrix per block = the 1 GB pass.
// Thread layout: c16 = t&3 selects a contiguous 16-channel (64 B) sub-row,
// jo = t>>2 selects j. Each thread issues 4 nontemporal b128 loads per j;
// score needs only a 4-lane (2-shuffle) reduction -> minimal DS pressure.
// ---------------------------------------------------------------------------
__global__ __launch_bounds__(256) void k1_score_mean_softmax(
    const float* __restrict__ matrix,   // [B,L,L,C]
    const float* __restrict__ Wp,       // [C]
    const float* __restrict__ bp,       // [1]
    float* __restrict__ attn,           // [B,L,L] softmax'd rows
    float* __restrict__ mmean)          // [B,L,C]
{
    const int bi  = blockIdx.x;         // b*1024 + i
    const int t   = threadIdx.x;        // 0..255
    const int c16 = t & 3;              // channel block (channels 16*c16..+15)
    const int jo  = t >> 2;             // 0..63, j offset within 64-j iteration
    const int cb  = c16 * 16;           // base channel

    __shared__ float s_score[kL];
    __shared__ float s_m[64 * 68];      // [j-group][channel], stride 68 (pad)
    __shared__ float s_red[16];

    const v4f* wpv = reinterpret_cast<const v4f*>(Wp);
    const v4f wp0 = wpv[c16 * 4 + 0];
    const v4f wp1 = wpv[c16 * 4 + 1];
    const v4f wp2 = wpv[c16 * 4 + 2];
    const v4f wp3 = wpv[c16 * 4 + 3];
    const float bpv = bp[0];
    const float* mbase = matrix + (size_t)bi * (kL * kC) + cb;

    v4f a0 = {}, a1 = {}, a2 = {}, a3 = {};

    #pragma unroll 2
    for (int it = 0; it < 16; ++it) {
        const int j = (it << 6) + jo;
        const v4f* rp = reinterpret_cast<const v4f*>(mbase + (size_t)j * kC);
        const v4f v0 = __builtin_nontemporal_load(rp + 0);
        const v4f v1 = __builtin_nontemporal_load(rp + 1);
        const v4f v2 = __builtin_nontemporal_load(rp + 2);
        const v4f v3 = __builtin_nontemporal_load(rp + 3);
        a0 += v0; a1 += v1; a2 += v2; a3 += v3;
        float p = v0.x * wp0.x + v0.y * wp0.y + v0.z * wp0.z + v0.w * wp0.w
                + v1.x * wp1.x + v1.y * wp1.y + v1.z * wp1.z + v1.w * wp1.w
                + v2.x * wp2.x + v2.y * wp2.y + v2.z * wp2.z + v2.w * wp2.w
                + v3.x * wp3.x + v3.y * wp3.y + v3.z * wp3.z + v3.w * wp3.w;
        // reduce over the 4 consecutive lanes sharing this j
        p += __shfl_xor(p, 1);
        p += __shfl_xor(p, 2);
        if (c16 == 0) s_score[j] = p + bpv;
    }

    // per-thread channel partials -> LDS matrix, then column sums
    {
        float* row = &s_m[jo * 68 + cb];
        row[ 0] = a0.x; row[ 1] = a0.y; row[ 2] = a0.z; row[ 3] = a0.w;
        row[ 4] = a1.x; row[ 5] = a1.y; row[ 6] = a1.z; row[ 7] = a1.w;
        row[ 8] = a2.x; row[ 9] = a2.y; row[10] = a2.z; row[11] = a2.w;
        row[12] = a3.x; row[13] = a3.y; row[14] = a3.z; row[15] = a3.w;
    }
    __syncthreads();

    if (t < kC) {
        float s = 0.f;
        #pragma unroll 8
        for (int r = 0; r < 64; ++r) s += s_m[r * 68 + t];
        mmean[(size_t)bi * kC + t] = s * (1.0f / (float)kL);
    }

    // ---- softmax over s_score[0..1023] ----
    float lmax = -3.402823466e38f;
    #pragma unroll
    for (int r = 0; r < 4; ++r) lmax = fmaxf(lmax, s_score[t + 256 * r]);
    #pragma unroll
    for (int off = 16; off > 0; off >>= 1) lmax = fmaxf(lmax, __shfl_xor(lmax, off));
    if ((t & 31) == 0) s_red[t >> 5] = lmax;
    __syncthreads();
    lmax = s_red[0];
    #pragma unroll
    for (int w = 1; w < 8; ++w) lmax = fmaxf(lmax, s_red[w]);

    float lsum = 0.f;
    #pragma unroll
    for (int r = 0; r < 4; ++r) {
        const int k = t + 256 * r;
        const float e = __expf(s_score[k] - lmax);
        s_score[k] = e;
        lsum += e;
    }
    #pragma unroll
    for (int off = 16; off > 0; off >>= 1) lsum += __shfl_xor(lsum, off);
    if ((t & 31) == 0) s_red[8 + (t >> 5)] = lsum;
    __syncthreads();
    lsum = 0.f;
    #pragma unroll
    for (int w = 0; w < 8; ++w) lsum += s_red[8 + w];
    const float inv = 1.0f / lsum;

    float* arow = attn + (size_t)bi * kL;
    #pragma unroll
    for (int r = 0; r < 4; ++r) {
        const int k = t + 256 * r;
        arow[k] = s_score[k] * inv;
    }
}

// ---------------------------------------------------------------------------
// Kernel 2: mapped[b,i,x] = sum_j attn[b,j,i] * seq[b,j,x]   (A^T * S)
// f32 WMMA 16x16x4. One wave per (b, i-tile, x-tile); 1024 waves total.
// attn (16 MB) was just written by k1 -> these reads hit the 192 MB L2.
// ---------------------------------------------------------------------------
__global__ __launch_bounds__(256) void k2_mapped_wmma(
    const float* __restrict__ attn,     // [B,L,L]  (row j, col i)
    const float* __restrict__ seq,      // [B,L,D]
    float* __restrict__ mapped)         // [B,L,D]
{
    const int wid  = (int)((blockIdx.x * blockDim.x + threadIdx.x) >> 5); // 0..1023
    const int lane = threadIdx.x & 31;
    const int nt   = wid & 3;           // x-tile (16 cols)
    const int it   = (wid >> 2) & 63;   // i-tile (16 rows)
    const int b    = wid >> 8;          // batch
    const int half = lane >> 4;         // K sub-group: {0,1} vs {2,3}
    const int m    = lane & 15;

    const int i0 = it * 16, x0 = nt * 16;
    const float* A = attn + (size_t)b * kL * kL;
    const float* S = seq  + (size_t)b * kL * kD;

    v8f acc = {};
    for (int k0 = 0; k0 < kL; k0 += 4) {
        const int r = k0 + 2 * half;
        v2f a, bb;
        a.x  = A[(size_t)r       * kL + i0 + m];   // attn[b, j=r,   i0+m]
        a.y  = A[(size_t)(r + 1) * kL + i0 + m];   // attn[b, j=r+1, i0+m]
        bb.x = S[(size_t)r       * kD + x0 + m];   // seq [b, j=r,   x0+m]
        bb.y = S[(size_t)(r + 1) * kD + x0 + m];
        acc = __builtin_amdgcn_wmma_f32_16x16x4_f32(false, a, false, bb,
                                                    (short)0, acc, false, false);
    }

    float* O = mapped + (size_t)b * kL * kD;
    #pragma unroll
    for (int v = 0; v < 8; ++v) {
        const int i = i0 + v + 8 * half;           // C/D layout: M = vgpr + 8*half
        O[(size_t)i * kD + x0 + m] = acc[v];
    }
}

// ---------------------------------------------------------------------------
// Kernel 3: out = concat([seq, mapped, mmean], -1) @ Wo + bo
// M = 4096, K = 192, N = 256. f32 WMMA 16x16x4; A gathered straight from the
// three sources (k-boundaries 64/128 are 4-aligned). One wave per
// (m-tile, 64-wide n-group) -> 4 accumulators; 1024 waves.
// ---------------------------------------------------------------------------
__global__ __launch_bounds__(256) void k3_out_wmma(
    const float* __restrict__ seq,      // [4096,64]
    const float* __restrict__ mapped,   // [4096,64]
    const float* __restrict__ mmean,    // [4096,64]
    const float* __restrict__ Wo,       // [192,256]
    const float* __restrict__ bo,       // [256]
    float* __restrict__ out)            // [4096,256]
{
    const int wid  = (int)((blockIdx.x * blockDim.x + threadIdx.x) >> 5); // 0..1023
    const int lane = threadIdx.x & 31;
    const int ng   = wid & 3;           // n-group of 64 columns
    const int mt   = wid >> 2;          // m-tile (16 rows), 0..255
    const int half = lane >> 4;
    const int m    = lane & 15;
    const int m0   = mt * 16;

    v8f acc0 = {}, acc1 = {}, acc2 = {}, acc3 = {};

    for (int k0 = 0; k0 < kK; k0 += 4) {
        const float* src;
        int kk;
        if (k0 < 64)       { src = seq;    kk = k0;       }
        else if (k0 < 128) { src = mapped; kk = k0 - 64;  }
        else               { src = mmean;  kk = k0 - 128; }

        v2f a;
        const float* rowp = src + (size_t)(m0 + m) * 64 + (kk + 2 * half);
        a.x = rowp[0];
        a.y = rowp[1];

        const int kr = k0 + 2 * half;
        const float* w0 = Wo + (size_t)kr * kOUT + ng * 64 + m;
        const float* w1 = w0 + kOUT;
        v2f b0, b1, b2, b3;
        b0.x = w0[0];  b0.y = w1[0];
        b1.x = w0[16]; b1.y = w1[16];
        b2.x = w0[32]; b2.y = w1[32];
        b3.x = w0[48]; b3.y = w1[48];

        acc0 = __builtin_amdgcn_wmma_f32_16x16x4_f32(false, a, false, b0, (short)0, acc0, false, false);
        acc1 = __builtin_amdgcn_wmma_f32_16x16x4_f32(false, a, false, b1, (short)0, acc1, false, false);
        acc2 = __builtin_amdgcn_wmma_f32_16x16x4_f32(false, a, false, b2, (short)0, acc2, false, false);
        acc3 = __builtin_amdgcn_wmma_f32_16x16x4_f32(false, a, false, b3, (short)0, acc3, false, false);
    }

    const int nb = ng * 64 + m;
    #pragma unroll
    for (int v = 0; v < 8; ++v) {
        const int row = m0 + v + 8 * half;
        float* orow = out + (size_t)row * kOUT;
        orow[nb +  0] = acc0[v] + bo[nb +  0];
        orow[nb + 16] = acc1[v] + bo[nb + 16];
        orow[nb + 32] = acc2[v] + bo[nb + 32];
        orow[nb + 48] = acc3[v] + bo[nb + 48];
    }
}

// ---------------------------------------------------------------------------
extern "C" void kernel_launch(void* const* d_in, const int* in_sizes, int n_in,
                              void* d_out, int out_size, void* d_ws, size_t ws_size,
                              hipStream_t stream) {
    const float* sequence = (const float*)d_in[0];  // [4,1024,64]
    const float* matrix   = (const float*)d_in[1];  // [4,1024,1024,64]
    const float* Wp       = (const float*)d_in[2];  // [64,1]
    const float* bp       = (const float*)d_in[3];  // [1]
    const float* Wo       = (const float*)d_in[4];  // [192,256]
    const float* bo       = (const float*)d_in[5];  // [256]
    float* out = (float*)d_out;                     // [4,1024,256]

    float* attn_ws   = (float*)d_ws;                          // 4*1024*1024 f32 (16 MB)
    float* mapped_ws = attn_ws + (size_t)kB * kL * kL;        // 4*1024*64
    float* mmean_ws  = mapped_ws + (size_t)kB * kL * kD;      // 4*1024*64

    k1_score_mean_softmax<<<kB * kL, 256, 0, stream>>>(matrix, Wp, bp, attn_ws, mmean_ws);
    k2_mapped_wmma<<<128, 256, 0, stream>>>(attn_ws, sequence, mapped_ws);
    k3_out_wmma<<<128, 256, 0, stream>>>(sequence, mapped_ws, mmean_ws, Wo, bo, out);
}